// DDRoPE_20925080666207
// MI455X (gfx1250) — compile-verified
//
#include <hip/hip_runtime.h>
#include <hip/hip_bf16.h>

typedef __attribute__((ext_vector_type(2))) float v2f;
typedef __attribute__((ext_vector_type(4))) float f4;
typedef __attribute__((ext_vector_type(8))) float v8f;

#define D_INPUT   2048
#define DD_PAIRS  256
#define NBATCH    4
#define T_LEN     4096
#define ROWS_TOTAL (NBATCH * T_LEN)   // 16384
#define CH        256                 // timesteps per cumsum chunk
#define NCHUNK    (T_LEN / CH)        // 16
#define KC        32                  // K-chunk for GEMM LDS staging
#define LDS_STRIDE 260                // 256 + 4 pad (bank-conflict avoidance)

// ---------------------------------------------------------------------------
// Kernel 0: transpose W (P x D) -> Wt (D x P) so B-fragment loads are coalesced
// ---------------------------------------------------------------------------
__global__ void k_transposeW(const float* __restrict__ W, float* __restrict__ Wt) {
    __shared__ float tile[16][17];
    const int bx = blockIdx.x;            // d-tile: 0..127
    const int by = blockIdx.y;            // p-tile: 0..15
    const int tx = threadIdx.x, ty = threadIdx.y;
    tile[ty][tx] = W[(size_t)(by * 16 + ty) * D_INPUT + bx * 16 + tx];
    __syncthreads();
    Wt[(size_t)(bx * 16 + ty) * DD_PAIRS + by * 16 + tx] = tile[tx][ty];
}

// ---------------------------------------------------------------------------
// Kernel 1: deltas[r][p] = sum_k x[r][k] * Wt[k][p] + bias[p]  via f32 WMMA
// Block = 256 threads (8 waves) covering 128 rows x 256 cols.
// Wave layout: 4 row-groups x 2 col-groups. Each wave: 2 M-strips (32 rows)
// x 8 N-tiles (128 cols) -> each B fragment feeds 2 WMMAs (halves DS loads).
// ---------------------------------------------------------------------------
__global__ void __launch_bounds__(256, 1)
k_gemm(const float* __restrict__ x, const float* __restrict__ Wt,
       const float* __restrict__ bias, float* __restrict__ deltas) {
    __shared__ float lds[KC * LDS_STRIDE];   // 33,280 B

    const int tid    = threadIdx.x;
    const int wid    = tid >> 5;
    const int lane   = tid & 31;
    const int half   = lane >> 4;            // 0: K 0,1 / 1: K 2,3 fragment half
    const int ln     = lane & 15;
    const int mgroup = wid & 3;              // 4 row groups of 32 rows
    const int cgroup = wid >> 2;             // 2 col groups of 128 cols
    const int rBase  = blockIdx.x * 128 + mgroup * 32;
    const int nBase  = cgroup * 128;

    v8f acc[2][8];
    #pragma unroll
    for (int m = 0; m < 2; ++m)
        #pragma unroll
        for (int nt = 0; nt < 8; ++nt) { v8f z = {}; acc[m][nt] = z; }

    // Per-lane A row pointers (ISA 16x4 f32 layout: lane<16 K=0,1; lane>=16 K=2,3)
    const float* arow0 = x + (size_t)(rBase + ln) * D_INPUT + half * 2;
    const float* arow1 = arow0 + (size_t)16 * D_INPUT;

    for (int kc = 0; kc < D_INPUT; kc += KC) {
        __syncthreads();
        // Cooperative stage of Wt[kc..kc+31][0..255] into LDS (float4, coalesced)
        #pragma unroll
        for (int j4 = 0; j4 < 8; ++j4) {
            const int idx = j4 * 256 + tid;       // float4 units, 0..2047
            const int row = idx >> 6;             // 64 float4 per row
            const int c4  = idx & 63;
            const f4 v = *(const f4*)(Wt + (size_t)(kc + row) * DD_PAIRS + c4 * 4);
            *(f4*)(&lds[row * LDS_STRIDE + c4 * 4]) = v;
        }
        __syncthreads();

        // A fragments for both M strips, 8 k-blocks of 4
        v2f afr0[8], afr1[8];
        #pragma unroll
        for (int kb = 0; kb < 8; ++kb) {
            afr0[kb] = *(const v2f*)(arow0 + kc + kb * 4);
            afr1[kb] = *(const v2f*)(arow1 + kc + kb * 4);
        }

        #pragma unroll
        for (int kb = 0; kb < 8; ++kb) {
            const int krow = kb * 4 + half * 2;
            const float* bp0 = &lds[krow * LDS_STRIDE + nBase + ln];
            #pragma unroll
            for (int nt = 0; nt < 8; ++nt) {
                v2f bfr;
                bfr.x = bp0[nt * 16];                // B[krow  ][n0+ln]
                bfr.y = bp0[LDS_STRIDE + nt * 16];   // B[krow+1][n0+ln]
                acc[0][nt] = __builtin_amdgcn_wmma_f32_16x16x4_f32(
                    false, afr0[kb], false, bfr, (short)0, acc[0][nt], false, false);
                acc[1][nt] = __builtin_amdgcn_wmma_f32_16x16x4_f32(
                    false, afr1[kb], false, bfr, (short)0, acc[1][nt], false, false);
            }
        }
    }

    // Epilogue: C layout -> VGPR v: lanes<16 row v, lanes>=16 row v+8
    #pragma unroll
    for (int m = 0; m < 2; ++m) {
        #pragma unroll
        for (int nt = 0; nt < 8; ++nt) {
            const int col = nBase + nt * 16 + ln;
            const float bv = bias[col];
            const int row0 = rBase + m * 16 + half * 8;
            #pragma unroll
            for (int v = 0; v < 8; ++v) {
                deltas[(size_t)(row0 + v) * DD_PAIRS + col] = acc[m][nt][v] + bv;
            }
        }
    }
}

// ---------------------------------------------------------------------------
// Kernel 2: per-(batch,chunk,pair) sum of deltas over CH timesteps
// ---------------------------------------------------------------------------
__global__ void k_chunksum(const float* __restrict__ deltas, float* __restrict__ csum) {
    const int blk = blockIdx.x;           // 0..63 = b*16 + c
    const int p = threadIdx.x;
    const int b = blk >> 4, c = blk & 15;
    size_t base = (size_t)(b * T_LEN + c * CH) * DD_PAIRS + p;
    float s = 0.f;
    for (int t = 0; t < CH; ++t) s += deltas[base + (size_t)t * DD_PAIRS];
    csum[blk * DD_PAIRS + p] = s;
}

// ---------------------------------------------------------------------------
// Kernel 3: exclusive scan of chunk sums per (batch, pair)
// ---------------------------------------------------------------------------
__global__ void k_chunkscan(const float* __restrict__ csum, float* __restrict__ cbase) {
    const int b = blockIdx.x;
    const int p = threadIdx.x;
    float run = 0.f;
    for (int c = 0; c < NCHUNK; ++c) {
        const int i = (b * NCHUNK + c) * DD_PAIRS + p;
        cbase[i] = run;
        run += csum[i];
    }
}

// ---------------------------------------------------------------------------
// Kernel 4: in-chunk inclusive cumsum + sincos + pairwise rotation
// ---------------------------------------------------------------------------
__global__ void k_rotate(const float* __restrict__ x, const float* __restrict__ deltas,
                         const float* __restrict__ cbase, float* __restrict__ out) {
    const int blk = blockIdx.x;           // b*16 + c
    const int p = threadIdx.x;
    const int b = blk >> 4, c = blk & 15;
    const int t0 = c * CH;
    size_t drow = (size_t)(b * T_LEN + t0) * DD_PAIRS + p;
    size_t xrow = (size_t)(b * T_LEN + t0) * D_INPUT + p;
    float angle = cbase[blk * DD_PAIRS + p];
    for (int t = 0; t < CH; ++t) {
        angle += deltas[drow + (size_t)t * DD_PAIRS];    // inclusive cumsum
        float sn, cs;
        __sincosf(angle, &sn, &cs);
        const float x1 = x[xrow + (size_t)t * D_INPUT];
        const float x2 = x[xrow + (size_t)t * D_INPUT + DD_PAIRS];
        out[xrow + (size_t)t * D_INPUT]            = x1 * cs - x2 * sn;
        out[xrow + (size_t)t * D_INPUT + DD_PAIRS] = x2 * cs + x1 * sn;
    }
}

// ---------------------------------------------------------------------------
// Kernel 5: copy channels [512, 2048) untouched (float4, grid-stride)
// ---------------------------------------------------------------------------
__global__ void k_copytail(const float* __restrict__ x, float* __restrict__ out) {
    const long long N4 = (long long)ROWS_TOTAL * 384;   // 1536 floats = 384 f4 per row
    long long i = (long long)blockIdx.x * blockDim.x + threadIdx.x;
    const long long stride = (long long)gridDim.x * blockDim.x;
    for (; i < N4; i += stride) {
        const long long row = i / 384;
        const long long c4  = i % 384;
        const long long off = row * 512 + 128 + c4;     // row stride 512 f4, skip 128 f4
        ((f4*)out)[off] = ((const f4*)x)[off];
    }
}

extern "C" void kernel_launch(void* const* d_in, const int* in_sizes, int n_in,
                              void* d_out, int out_size, void* d_ws, size_t ws_size,
                              hipStream_t stream) {
    (void)in_sizes; (void)n_in; (void)out_size; (void)ws_size;
    const float* x    = (const float*)d_in[0];   // (4,4096,2048) f32
    const float* W    = (const float*)d_in[1];   // (256,2048) f32
    const float* bias = (const float*)d_in[2];   // (256,) f32
    float* out = (float*)d_out;

    // workspace layout (floats): Wt | deltas | csum | cbase  (~19 MB)
    float* Wt     = (float*)d_ws;                          // 2048*256
    float* deltas = Wt + (size_t)D_INPUT * DD_PAIRS;       // 16384*256
    float* csum   = deltas + (size_t)ROWS_TOTAL * DD_PAIRS;// 64*256
    float* cbase  = csum + (size_t)NBATCH * NCHUNK * DD_PAIRS;

    k_transposeW<<<dim3(D_INPUT / 16, DD_PAIRS / 16), dim3(16, 16), 0, stream>>>(W, Wt);
    k_gemm      <<<ROWS_TOTAL / 128, 256, 0, stream>>>(x, Wt, bias, deltas);
    k_chunksum  <<<NBATCH * NCHUNK, DD_PAIRS, 0, stream>>>(deltas, csum);
    k_chunkscan <<<NBATCH, DD_PAIRS, 0, stream>>>(csum, cbase);
    k_rotate    <<<NBATCH * NCHUNK, DD_PAIRS, 0, stream>>>(x, deltas, cbase, out);
    k_copytail  <<<2048, 256, 0, stream>>>(x, out);
}